// FarChamferDist_24790551233439
// MI455X (gfx1250) — compile-verified
//
#include <hip/hip_runtime.h>

typedef __attribute__((ext_vector_type(2))) float v2f;
typedef __attribute__((ext_vector_type(8))) float v8f;

namespace {
constexpr int   kB            = 8;
constexpr int   kK1           = 2048;   // adv points per batch
constexpr int   kK2           = 8192;   // ori points per batch
constexpr int   kNumAdd       = 16;
constexpr int   kClusterP     = kK1 / kNumAdd;  // 128
constexpr float kCdW          = 0.1f;
constexpr float kEps          = 1e-7f;
constexpr int   kTilesPerWave = 4;                 // adv 16-row tiles per wave
constexpr int   kATiles       = kK1 / 16;          // 128
constexpr int   kWavesPerB    = kATiles / kTilesPerWave;  // 32
constexpr int   kChamferWaves = kB * kWavesPerB;   // 256
constexpr int   kOTiles       = kK2 / 16;          // 512
constexpr int   kFarBlocks    = kB * kNumAdd;      // 128
}

// Raw v_min_num_f32 acc = min(acc, x): avoids the v_max_num canonicalize the
// compiler otherwise inserts before minnum on WMMA intrinsic outputs.
__device__ __forceinline__ void vmin_acc(float& acc, float x) {
  asm("v_min_num_f32 %0, %0, %1" : "+v"(acc) : "v"(x));
}

// One wave (32 lanes) owns 4 adv 16-point tiles of one batch and scans all
// 8192 ori points in 16-point tiles. Each ori tile -> 4x V_WMMA_F32_16X16X4_F32
// producing d2' = |o|^2 - 2 a.o for a full 16x16 pair block; 8 v_min per WMMA
// accumulates (the minimum possible). After the scan: xor-shuffle min across
// the 16 columns, add |a|^2 per row, clamp, sum rows -> one partial per wave.
__global__ __launch_bounds__(32) void chamfer_wmma_kernel(
    const float* __restrict__ adv, const float* __restrict__ ori,
    float* __restrict__ ws_ch) {
  const int wave   = blockIdx.x;
  const int b      = wave / kWavesPerB;
  const int atile0 = (wave % kWavesPerB) * kTilesPerWave;
  const int lane   = threadIdx.x;
  const int m16    = lane & 15;
  const bool hi    = lane >= 16;

  const float* advb = adv + (size_t)b * kK1 * 3;
  const float* orib = ori + (size_t)b * kK2 * 3;

  // A tiles: 16x4 f32. Lanes 0-15: VGPR0=K0(-2ax), VGPR1=K1(-2ay).
  //          Lanes 16-31: VGPR0=K2(-2az), VGPR1=K3(1.0).
  v2f Amat[kTilesPerWave];
  float a2l[kTilesPerWave];   // |a|^2 of point (row m16) of each tile
#pragma unroll
  for (int t = 0; t < kTilesPerWave; ++t) {
    const int p   = (atile0 + t) * 16 + m16;
    const float ax = advb[p * 3 + 0];
    const float ay = advb[p * 3 + 1];
    const float az = advb[p * 3 + 2];
    a2l[t] = ax * ax + ay * ay + az * az;
    Amat[t].x = hi ? (-2.0f * az) : (-2.0f * ax);
    Amat[t].y = hi ? 1.0f         : (-2.0f * ay);
  }

  float mn[kTilesPerWave][8];
#pragma unroll
  for (int t = 0; t < kTilesPerWave; ++t)
#pragma unroll
    for (int i = 0; i < 8; ++i) mn[t][i] = 3.0e38f;

  for (int ot = 0; ot < kOTiles; ++ot) {
    const int p    = ot * 16 + m16;
    const float ox = orib[p * 3 + 0];
    const float oy = orib[p * 3 + 1];
    const float oz = orib[p * 3 + 2];
    const float o2 = ox * ox + oy * oy + oz * oz;
    // B tile: 4x16 f32. Lanes 0-15: VGPR0=K0(ox), VGPR1=K1(oy).
    //         Lanes 16-31: VGPR0=K2(oz), VGPR1=K3(|o|^2).
    v2f Bmat;
    Bmat.x = hi ? oz : ox;
    Bmat.y = hi ? o2 : oy;
#pragma unroll
    for (int t = 0; t < kTilesPerWave; ++t) {
      v8f c = {0.f, 0.f, 0.f, 0.f, 0.f, 0.f, 0.f, 0.f};
      v8f d = __builtin_amdgcn_wmma_f32_16x16x4_f32(
          /*neg_a=*/false, Amat[t], /*neg_b=*/false, Bmat,
          /*c_mod=*/(short)0, c, /*reuse_a=*/false, /*reuse_b=*/false);
#pragma unroll
      for (int i = 0; i < 8; ++i) vmin_acc(mn[t][i], d[i]);
    }
  }

  // Column (N) reduction inside each 16-lane half, then add |a|^2 and clamp.
  float s = 0.0f;
#pragma unroll
  for (int t = 0; t < kTilesPerWave; ++t) {
#pragma unroll
    for (int i = 0; i < 8; ++i) {
      float x = mn[t][i];
      x = fminf(x, __shfl_xor(x, 1, 32));
      x = fminf(x, __shfl_xor(x, 2, 32));
      x = fminf(x, __shfl_xor(x, 4, 32));
      x = fminf(x, __shfl_xor(x, 8, 32));
      // row index for component i is i (lower half) or i+8 (upper half);
      // lane r (< 16) holds a2 of row r.
      const float a2r = __shfl(a2l[t], i + (hi ? 8 : 0), 32);
      s += fmaxf(x + a2r, 0.0f);
    }
  }
  s += __shfl_xor(s, 16, 32);   // combine rows 0-7 with rows 8-15
  if (lane == 0) ws_ch[wave] = s;
}

// One block per (batch, cluster): 128 points in LDS, each thread takes max_j
// ||p_j - p_i + eps||, block max-reduce, write raw cluster max to d_ws.
__global__ __launch_bounds__(128) void farthest_kernel(
    const float* __restrict__ adv, float* __restrict__ ws_far) {
  __shared__ float sx[kClusterP], sy[kClusterP], sz[kClusterP];
  __shared__ float red[4];
  const int blk = blockIdx.x;
  const int b   = blk / kNumAdd;
  const int c   = blk % kNumAdd;
  const int tid = threadIdx.x;

  const float* base = adv + ((size_t)b * kK1 + (size_t)c * kClusterP) * 3;
  sx[tid] = base[tid * 3 + 0];
  sy[tid] = base[tid * 3 + 1];
  sz[tid] = base[tid * 3 + 2];
  __syncthreads();

  const float px = sx[tid], py = sy[tid], pz = sz[tid];
  float mx = 0.0f;
#pragma unroll 4
  for (int j = 0; j < kClusterP; ++j) {
    const float dx = sx[j] - px + kEps;
    const float dy = sy[j] - py + kEps;
    const float dz = sz[j] - pz + kEps;
    mx = fmaxf(mx, sqrtf(dx * dx + dy * dy + dz * dz));
  }
  mx = fmaxf(mx, __shfl_xor(mx, 1, 32));
  mx = fmaxf(mx, __shfl_xor(mx, 2, 32));
  mx = fmaxf(mx, __shfl_xor(mx, 4, 32));
  mx = fmaxf(mx, __shfl_xor(mx, 8, 32));
  mx = fmaxf(mx, __shfl_xor(mx, 16, 32));
  if ((tid & 31) == 0) red[tid >> 5] = mx;
  __syncthreads();
  if (tid == 0)
    ws_far[blk] = fmaxf(fmaxf(red[0], red[1]), fmaxf(red[2], red[3]));
}

// Deterministic single-thread combine (no float atomics -> replay-stable).
__global__ void finalize_kernel(const float* __restrict__ ws_ch,
                                const float* __restrict__ ws_far,
                                const float* __restrict__ weights,
                                float* __restrict__ out) {
  if (threadIdx.x == 0 && blockIdx.x == 0) {
    float acc = 0.0f;
    for (int b = 0; b < kB; ++b) {
      float ch = 0.0f;
      for (int w = 0; w < kWavesPerB; ++w) ch += ws_ch[b * kWavesPerB + w];
      float fa = 0.0f;
      for (int c = 0; c < kNumAdd; ++c) fa += ws_far[b * kNumAdd + c];
      acc += weights[b] * (fa + kCdW * (ch / (float)kK1));
    }
    out[0] = acc / (float)kB;
  }
}

extern "C" void kernel_launch(void* const* d_in, const int* in_sizes, int n_in,
                              void* d_out, int out_size, void* d_ws, size_t ws_size,
                              hipStream_t stream) {
  const float* adv     = (const float*)d_in[0];  // [8,2048,3]
  const float* ori     = (const float*)d_in[1];  // [8,8192,3]
  const float* weights = (const float*)d_in[2];  // [8]
  float* ws_ch  = (float*)d_ws;                  // 256 partials
  float* ws_far = ws_ch + kChamferWaves;         // 128 partials

  chamfer_wmma_kernel<<<kChamferWaves, 32, 0, stream>>>(adv, ori, ws_ch);
  farthest_kernel<<<kFarBlocks, kClusterP, 0, stream>>>(adv, ws_far);
  finalize_kernel<<<1, 32, 0, stream>>>(ws_ch, ws_far, weights, (float*)d_out);
}